// TextCrossAttention_43241730736703
// MI455X (gfx1250) — compile-verified
//
#include <hip/hip_runtime.h>
#include <stdint.h>

typedef __attribute__((ext_vector_type(2))) float v2f;
typedef __attribute__((ext_vector_type(4))) float v4f;
typedef __attribute__((ext_vector_type(8))) float v8f;
typedef int v4i __attribute__((vector_size(16)));

#define BATCH 64
#define CHN   512
#define TDIM  256
#define NH    4
#define DHEAD 64
#define TLAT  1000
#define TLATP 1008
#define TTXT  512
#define ATT_SCALE 0.0625f   // 256^-0.5

static __device__ __forceinline__ v8f wmma4(v2f a, v2f b, v8f c) {
  // D = A(16x4) * B(4x16) + C, f32
  return __builtin_amdgcn_wmma_f32_16x16x4_f32(false, a, false, b, (short)0, c,
                                               false, false);
}

#if __has_builtin(__builtin_amdgcn_global_load_async_to_lds_b128) && \
    __has_builtin(__builtin_amdgcn_s_wait_asynccnt)
#define USE_ASYNC 1
#else
#define USE_ASYNC 0
#endif

// copy 16 bytes global -> LDS (async when available)
static __device__ __forceinline__ void copy16(const float* g, float* l) {
#if USE_ASYNC
  __attribute__((address_space(1))) v4i* gp =
      (__attribute__((address_space(1))) v4i*)(unsigned long long)(uintptr_t)g;
  __attribute__((address_space(3))) v4i* lp =
      (__attribute__((address_space(3))) v4i*)(unsigned int)(uintptr_t)l;
  __builtin_amdgcn_global_load_async_to_lds_b128(gp, lp, 0, 0);
#else
  v4f v = *(const v4f*)g;
  *(v4f*)l = v;
#endif
}

static __device__ __forceinline__ void wait_async_le2() {
#if USE_ASYNC
  __builtin_amdgcn_s_wait_asynccnt(2);
#endif
}
static __device__ __forceinline__ void wait_async_0() {
#if USE_ASYNC
  __builtin_amdgcn_s_wait_asynccnt(0);
#endif
}

// ---------------------------------------------------------------------------
// Projection + bias + epilogue (RoPE / plain / transposed-store).
//  X: (B, Cin, T)   W: (256, Cin)   bias: (256)
//  mode 0: out (B*NH, Tpad, 64)
//  mode 1: same + RoPE
//  mode 2: out (B*NH, 64, Tpad)   (transposed, for V)
// Workgroup: 8 waves, each wave a 32(d)x32(t) 2x2 WMMA block -> 64d x 128t.
// ---------------------------------------------------------------------------
__global__ __launch_bounds__(256)
void proj_rope_kernel(const float* __restrict__ X, const float* __restrict__ W,
                      const float* __restrict__ bias,
                      const float* __restrict__ theta,
                      float* __restrict__ out,
                      int Cin, int T, int Tpad, float inv_seq, int mode)
{
  __shared__ float tile[64][129];         // [d_local][t_local], padded
  const int b     = blockIdx.z;
  const int h     = blockIdx.y;
  const int t0    = blockIdx.x * 128;
  const int wave  = threadIdx.x >> 5;
  const int lane  = threadIdx.x & 31;
  const int ln    = lane & 15;
  const int kb    = (lane >> 4) << 1;     // 0 or 2
  const int mbase = (lane >> 4) << 3;     // 0 or 8
  const int mt    = (wave & 1) << 5;      // d offset in head: 0/32
  const int nt    = (wave >> 1) << 5;     // t offset in block: 0/32/64/96
  const int d0    = h * DHEAD + mt;

  const int  ttA = t0 + nt + ln;
  const int  ttB = ttA + 16;
  const bool tvA = (ttA < T), tvB = (ttB < T);
  const float* __restrict__ W0 = W + (size_t)(d0 + ln) * Cin;
  const float* __restrict__ W1 = W0 + (size_t)16 * Cin;
  const float* __restrict__ XA = X + (size_t)b * Cin * T + (tvA ? ttA : 0);
  const float* __restrict__ XB = X + (size_t)b * Cin * T + (tvB ? ttB : 0);

  v8f a00 = {}, a01 = {}, a10 = {}, a11 = {};
#pragma unroll 2
  for (int c = 0; c < Cin; c += 4) {
    v2f wa, wb, xA, xB;
    wa.x = W0[c + kb]; wa.y = W0[c + kb + 1];
    wb.x = W1[c + kb]; wb.y = W1[c + kb + 1];
    float fa0 = XA[(size_t)(c + kb) * T];
    float fa1 = XA[(size_t)(c + kb + 1) * T];
    float fb0 = XB[(size_t)(c + kb) * T];
    float fb1 = XB[(size_t)(c + kb + 1) * T];
    xA.x = tvA ? fa0 : 0.0f;  xA.y = tvA ? fa1 : 0.0f;
    xB.x = tvB ? fb0 : 0.0f;  xB.y = tvB ? fb1 : 0.0f;
    a00 = wmma4(wa, xA, a00);
    a01 = wmma4(wa, xB, a01);
    a10 = wmma4(wb, xA, a10);
    a11 = wmma4(wb, xB, a11);
  }

  const int tlA = nt + ln, tlB = tlA + 16;
#pragma unroll
  for (int r = 0; r < 8; ++r) {
    int m = r + mbase;
    float b0v = bias[d0 + m];
    float b1v = bias[d0 + 16 + m];
    tile[mt + m][tlA]      = a00[r] + b0v;
    tile[mt + m][tlB]      = a01[r] + b0v;
    tile[mt + 16 + m][tlA] = a10[r] + b1v;
    tile[mt + 16 + m][tlB] = a11[r] + b1v;
  }
  __syncthreads();

  const size_t bh = (size_t)(b * NH + h);
  if (mode == 2) {
    // transposed store: out[bh][dl][t], t fastest for coalescing
#pragma unroll
    for (int i = 0; i < 32; ++i) {
      int idx = threadIdx.x + i * 256;   // 64 x 128
      int tl2 = idx & 127;
      int dl  = idx >> 7;
      int t   = t0 + tl2;
      if (t >= Tpad) continue;
      out[(bh * DHEAD + dl) * Tpad + t] = tile[dl][tl2];
    }
  } else {
#pragma unroll
    for (int i = 0; i < 32; ++i) {
      int idx = threadIdx.x + i * 256;
      int dl  = idx & 63;
      int tl2 = idx >> 6;
      int t   = t0 + tl2;
      if (t >= Tpad) continue;
      float val;
      if (mode == 1) {
        int   j  = dl & 31;
        float x1 = tile[j][tl2];
        float x2 = tile[j + 32][tl2];
        float ang = ((float)t * inv_seq) * theta[j];
        float sn, cs;
        __sincosf(ang, &sn, &cs);
        val = (dl < 32) ? (x1 * cs - x2 * sn) : (x1 * sn + x2 * cs);
      } else {
        val = tile[dl][tl2];
      }
      out[(bh * Tpad + t) * DHEAD + dl] = val;
    }
  }
}

// ---------------------------------------------------------------------------
// Fused masked flash attention, block-synchronous with double-buffered
// async K/Vt tile staging in LDS (all 8 waves share (b,h)).
//  Q: (B*NH, TLATP, 64)  K: (B*NH, 512, 64)  Vt: (B*NH, 64, 512)
//  O: (B, TLAT, 256)
// ---------------------------------------------------------------------------
#define KSTR 68   // K tile row stride (words): 16B-aligned rows, conflict-free
#define VSTR 20   // Vt tile row stride (words)

__global__ __launch_bounds__(256)
void attn_kernel(const float* __restrict__ Q, const float* __restrict__ K,
                 const float* __restrict__ Vt, const int* __restrict__ tmask,
                 float* __restrict__ O)
{
  __shared__ float maskbias[TTXT];
  __shared__ float pbuf[8][16 * 17];
  __shared__ __attribute__((aligned(16))) float kbuf[2][16 * KSTR];
  __shared__ __attribute__((aligned(16))) float vbuf[2][64 * VSTR];

  const int b    = blockIdx.z;
  const int h    = blockIdx.y;
  const int tid  = threadIdx.x;
  const int wave = tid >> 5;
  const int lane = tid & 31;
  const int ln   = lane & 15;
  const int hi   = lane >> 4;
  const int kb   = hi << 1;

  for (int i = tid; i < TTXT; i += 256)
    maskbias[i] = tmask[b * TTXT + i] ? 0.0f : -1e30f;

  const int  qt_raw = blockIdx.x * 8 + wave;
  const bool qvalid = (qt_raw * 16 < TLATP);
  const int  q0     = (qvalid ? qt_raw : 62) * 16;

  const size_t bh = (size_t)(b * NH + h);
  const float* __restrict__ Qrow   = Q + (bh * TLATP + q0 + ln) * DHEAD;
  const float* __restrict__ Kbase  = K + bh * TTXT * DHEAD;
  const float* __restrict__ Vtbase = Vt + bh * DHEAD * TTXT;
  float* pw = &pbuf[wave][0];

  // cooperative tile-copy assignments (one b128 per thread per tile each)
  const int krow = tid >> 4;             // 0..15
  const int kcol = (tid & 15) << 2;      // 0..60
  const int vrow = tid >> 2;             // 0..63
  const int vcol = (tid & 3) << 2;       // 0,4,8,12
  const float* Kg = Kbase + (size_t)krow * DHEAD + kcol;
  const float* Vg = Vtbase + (size_t)vrow * TTXT + vcol;
  float* Kl0 = &kbuf[0][krow * KSTR + kcol];
  float* Vl0 = &vbuf[0][vrow * VSTR + vcol];
  float* Kl1 = &kbuf[1][krow * KSTR + kcol];
  float* Vl1 = &vbuf[1][vrow * VSTR + vcol];

  // hoist this wave's Q operand (whole dh range) into registers
  v2f qreg[16];
#pragma unroll
  for (int c4 = 0; c4 < 16; ++c4) {
    qreg[c4].x = Qrow[c4 * 4 + kb];
    qreg[c4].y = Qrow[c4 * 4 + kb + 1];
  }

  v8f o0 = {}, o1 = {}, o2 = {}, o3 = {};
  float m_run = -3.0e38f;
  float l_run = 0.0f;

  // prologue: stage tile 0 into buffer 0
  copy16(Kg, Kl0);
  copy16(Vg, Vl0);

  for (int kt = 0; kt < TTXT / 16; ++kt) {
    const int k0  = kt * 16;
    const int cur = kt & 1;

    if (kt + 1 < TTXT / 16) {        // prefetch next tile into other buffer
      const size_t knext = (size_t)(k0 + 16);
      copy16(Kg + knext * DHEAD, cur ? Kl0 : Kl1);
      copy16(Vg + knext,         cur ? Vl0 : Vl1);
      wait_async_le2();              // tile kt's 2 loads complete (in-order)
    } else {
      wait_async_0();
    }
    __syncthreads();                 // tile kt visible block-wide

    const float* kcur = &kbuf[cur][0];
    const float* vcur = &vbuf[cur][0];

    // S^T tile: A = K rows from LDS (M=k), B = Q (registers)
    v8f s = {};
#pragma unroll
    for (int c4 = 0; c4 < 16; ++c4) {
      v2f a;
      a.x = kcur[ln * KSTR + c4 * 4 + kb];
      a.y = kcur[ln * KSTR + c4 * 4 + kb + 1];
      s = wmma4(a, qreg[c4], s);
    }

    // scale + additive key mask; online softmax bookkeeping
    float p[8];
    float mloc = -3.0e38f;
#pragma unroll
    for (int r = 0; r < 8; ++r) {
      float sv = s[r] * ATT_SCALE + maskbias[k0 + r + 8 * hi];
      p[r] = sv;
      mloc = fmaxf(mloc, sv);
    }
    mloc = fmaxf(mloc, __shfl_xor(mloc, 16, 32));
    float newm = fmaxf(m_run, mloc);
    float fac  = __expf(m_run - newm);
    m_run = newm;
    l_run *= fac;
    o0 = o0 * fac; o1 = o1 * fac; o2 = o2 * fac; o3 = o3 * fac;

#pragma unroll
    for (int r = 0; r < 8; ++r) {
      float pv = __expf(p[r] - newm);
      p[r] = pv;
      l_run += pv;
    }
#pragma unroll
    for (int r = 0; r < 8; ++r)
      pw[(r + 8 * hi) * 17 + ln] = p[r];

    // O^T += Vt * P^T : A = Vt rows from LDS (M=dh), B = P from LDS (N=q)
#pragma unroll
    for (int c4 = 0; c4 < 4; ++c4) {
      const int col = c4 * 4 + kb;
      v2f bb;
      bb.x = pw[col * 17 + ln];
      bb.y = pw[(col + 1) * 17 + ln];
      v2f a0; a0.x = vcur[(0  + ln) * VSTR + col]; a0.y = vcur[(0  + ln) * VSTR + col + 1];
      o0 = wmma4(a0, bb, o0);
      v2f a1; a1.x = vcur[(16 + ln) * VSTR + col]; a1.y = vcur[(16 + ln) * VSTR + col + 1];
      o1 = wmma4(a1, bb, o1);
      v2f a2; a2.x = vcur[(32 + ln) * VSTR + col]; a2.y = vcur[(32 + ln) * VSTR + col + 1];
      o2 = wmma4(a2, bb, o2);
      v2f a3; a3.x = vcur[(48 + ln) * VSTR + col]; a3.y = vcur[(48 + ln) * VSTR + col + 1];
      o3 = wmma4(a3, bb, o3);
    }
    __syncthreads();                 // done reading buf[cur] before reuse
  }

  float l_tot = l_run + __shfl_xor(l_run, 16, 32);
  float inv_l = 1.0f / l_tot;

  const int t = q0 + ln;             // lane's q column == output row t
  if (qvalid && t < TLAT) {
    float* orow = O + ((size_t)b * TLAT + t) * TDIM + h * DHEAD + 8 * hi;
#pragma unroll
    for (int r = 0; r < 8; ++r) {
      orow[r]      = o0[r] * inv_l;
      orow[16 + r] = o1[r] * inv_l;
      orow[32 + r] = o2[r] * inv_l;
      orow[48 + r] = o3[r] * inv_l;
    }
  }
}

// ---------------------------------------------------------------------------
// Output projection + bias + residual + latent mask.  2x2 blocking per wave.
//  AO: (B, TLAT, 256)  Wo: (512, 256)  out/X: (B, 512, TLAT)
// Workgroup: 8 waves (2 ch-groups x 4 t-groups) -> 64ch x 128t per block.
// ---------------------------------------------------------------------------
__global__ __launch_bounds__(256)
void outproj_kernel(const float* __restrict__ AO, const float* __restrict__ Wo,
                    const float* __restrict__ bo, const float* __restrict__ X,
                    const float* __restrict__ latmask, float* __restrict__ out)
{
  const int b    = blockIdx.z;
  const int wave = threadIdx.x >> 5;
  const int lane = threadIdx.x & 31;
  const int ln   = lane & 15;
  const int hi   = lane >> 4;
  const int kb   = hi << 1;
  const int ch0  = blockIdx.y * 64 + ((wave & 1) << 5);
  const int tb   = blockIdx.x * 128 + ((wave >> 1) << 5);

  const int  tA = tb + ln, tB = tA + 16;
  const bool tvA = (tA < TLAT), tvB = (tB < TLAT);
  const float* __restrict__ A0 = Wo + (size_t)(ch0 + ln) * TDIM;
  const float* __restrict__ A1 = A0 + (size_t)16 * TDIM;
  const float* __restrict__ B0 = AO + ((size_t)b * TLAT + (tvA ? tA : TLAT - 1)) * TDIM;
  const float* __restrict__ B1 = AO + ((size_t)b * TLAT + (tvB ? tB : TLAT - 1)) * TDIM;

  v8f a00 = {}, a01 = {}, a10 = {}, a11 = {};
#pragma unroll 2
  for (int c = 0; c < TDIM; c += 4) {
    v2f wa, wb, xA, xB;
    wa.x = A0[c + kb]; wa.y = A0[c + kb + 1];
    wb.x = A1[c + kb]; wb.y = A1[c + kb + 1];
    xA.x = B0[c + kb]; xA.y = B0[c + kb + 1];
    xB.x = B1[c + kb]; xB.y = B1[c + kb + 1];
    a00 = wmma4(wa, xA, a00);
    a01 = wmma4(wa, xB, a01);
    a10 = wmma4(wb, xA, a10);
    a11 = wmma4(wb, xB, a11);
  }

  const float lmA = tvA ? latmask[(size_t)b * TLAT + tA] : 0.0f;
  const float lmB = tvB ? latmask[(size_t)b * TLAT + tB] : 0.0f;
#pragma unroll
  for (int r = 0; r < 8; ++r) {
    int chA = ch0 + r + 8 * hi;
    int chB = chA + 16;
    float boA = bo[chA], boB = bo[chB];
    if (tvA) {
      size_t i0 = ((size_t)b * CHN + chA) * TLAT + tA;
      size_t i1 = ((size_t)b * CHN + chB) * TLAT + tA;
      out[i0] = (X[i0] + a00[r] + boA) * lmA;
      out[i1] = (X[i1] + a10[r] + boB) * lmA;
    }
    if (tvB) {
      size_t i0 = ((size_t)b * CHN + chA) * TLAT + tB;
      size_t i1 = ((size_t)b * CHN + chB) * TLAT + tB;
      out[i0] = (X[i0] + a01[r] + boA) * lmB;
      out[i1] = (X[i1] + a11[r] + boB) * lmB;
    }
  }
}

// ---------------------------------------------------------------------------
extern "C" void kernel_launch(void* const* d_in, const int* in_sizes, int n_in,
                              void* d_out, int out_size, void* d_ws, size_t ws_size,
                              hipStream_t stream)
{
  const float* x       = (const float*)d_in[0];
  const float* text    = (const float*)d_in[1];
  const float* latmask = (const float*)d_in[2];
  const int*   tmask   = (const int*)d_in[3];
  const float* Wq      = (const float*)d_in[4];
  const float* bq      = (const float*)d_in[5];
  const float* Wk      = (const float*)d_in[6];
  const float* bk      = (const float*)d_in[7];
  const float* Wv      = (const float*)d_in[8];
  const float* bv      = (const float*)d_in[9];
  const float* Wo      = (const float*)d_in[10];
  const float* bo      = (const float*)d_in[11];
  const float* theta   = (const float*)d_in[12];

  float* Qb  = (float*)d_ws;                                 // B*NH*1008*64
  float* Kb  = Qb + (size_t)BATCH * NH * TLATP * DHEAD;      // B*NH*512*64
  float* Vtb = Kb + (size_t)BATCH * NH * TTXT * DHEAD;       // B*NH*64*512
  float* AO  = Vtb + (size_t)BATCH * NH * DHEAD * TTXT;      // B*1000*256

  dim3 blk(256);
  // Q = x * Wq^T + bq, RoPE(seq=1000)
  proj_rope_kernel<<<dim3(8, NH, BATCH), blk, 0, stream>>>(
      x, Wq, bq, theta, Qb, CHN, TLAT, TLATP, 1.0f / (float)TLAT, 1);
  // K = text * Wk^T + bk, RoPE(seq=512)
  proj_rope_kernel<<<dim3(4, NH, BATCH), blk, 0, stream>>>(
      text, Wk, bk, theta, Kb, TDIM, TTXT, TTXT, 1.0f / (float)TTXT, 1);
  // V = text * Wv^T + bv, stored transposed
  proj_rope_kernel<<<dim3(4, NH, BATCH), blk, 0, stream>>>(
      text, Wv, bv, theta, Vtb, TDIM, TTXT, TTXT, 0.0f, 2);
  // fused masked softmax attention (async double-buffered K/Vt staging)
  attn_kernel<<<dim3(8, NH, BATCH), blk, 0, stream>>>(Qb, Kb, Vtb, tmask, AO);
  // output projection + residual + mask
  outproj_kernel<<<dim3(8, 8, BATCH), blk, 0, stream>>>(AO, Wo, bo, x, latmask,
                                                        (float*)d_out);
}